// GIN_13657996001651
// MI455X (gfx1250) — compile-verified
//
#include <hip/hip_runtime.h>
#include <hip/hip_bf16.h>

// Problem constants (match reference)
#define NN 50000
#define EE 800000
#define DD 128
#define LL 3
#define RR 3
#define CC 16
#define BN_EPS 1e-5f
#define COMMIT_W 0.25f

#define AP 132  // padded LDS pitch (floats): stride%64 == 4 -> conflict-free fragment reads

typedef float v2f __attribute__((ext_vector_type(2)));
typedef float v8f __attribute__((ext_vector_type(8)));

// ---------------- utility kernels ----------------

__global__ void zero_f32(float* __restrict__ p, long long count) {
    long long i = (long long)blockIdx.x * blockDim.x + threadIdx.x;
    if (i < count) p[i] = 0.0f;
}

// ---------------- edge scatter: neigh[dst] += h[src] ----------------
__global__ void scatter_add_kernel(const float* __restrict__ h,
                                   const int* __restrict__ src,
                                   const int* __restrict__ dst,
                                   float* __restrict__ neigh) {
    long long t = (long long)blockIdx.x * blockDim.x + threadIdx.x;
    int e = (int)(t >> 7);        // t / 128
    int d = (int)(t & 127);       // t % 128
    if (e < EE) {
        float v = h[(size_t)src[e] * DD + d];
        atomicAdd(&neigh[(size_t)dst[e] * DD + d], v);
    }
}

// ---------------- W transpose: WT[l][n][k] = W[l][k][n] ----------------
// grid: (DD*DD/256, L) ; one thread per element
__global__ void transpose_w(const float* __restrict__ W, float* __restrict__ WT) {
    int l = blockIdx.y;
    int idx = blockIdx.x * 256 + threadIdx.x;   // 0..16383
    int k = idx >> 7;
    int n = idx & 127;
    WT[(size_t)l * DD * DD + (size_t)n * DD + k] =
        W[(size_t)l * DD * DD + (size_t)k * DD + n];
}

// ---------------- codebook row L2-normalize ----------------
__global__ void norm_codebooks(const float* __restrict__ cb, float* __restrict__ cbn) {
    int row = blockIdx.x;
    int d = threadIdx.x;
    __shared__ float sred[DD];
    float v = cb[(size_t)row * DD + d];
    sred[d] = v * v;
    __syncthreads();
    for (int off = 64; off > 0; off >>= 1) {
        if (d < off) sred[d] += sred[d + off];
        __syncthreads();
    }
    float nrm = fmaxf(sqrtf(sred[0]), 1e-12f);
    cbn[(size_t)row * DD + d] = v / nrm;
}

// ---------------- WMMA GEMM + fused A-construction + BN stats ----------------
// Z = A @ W + bias
//   MODE 0: A[m,k] = A0[m,k] + A1[m,k]                 (h + neigh)
//   MODE 1: A[m,k] = relu(scale[k]*A0[m,k] + shift[k]) (BN1 affine + ReLU)
// Block = 256 thr = 8 waves; wave w -> 16x16 tile (rows blockIdx.x*16, cols w*16).
// A tile staged into LDS (fused construction, float4 loads, padded pitch).
// B fragments preloaded to registers from pre-transposed WT (contiguous v2f).
// Inner loop: ds_load_b64(A) + V_WMMA_F32_16X16X4_F32, fully unrolled (32x).
template <int MODE>
__global__ void __launch_bounds__(256) gemm_bn_stats(
        const float* __restrict__ A0,
        const float* __restrict__ A1,
        const float* __restrict__ scale,
        const float* __restrict__ shift,
        const float* __restrict__ WT,      // [DD][DD] transposed weights
        const float* __restrict__ bias,
        float* __restrict__ Z,
        float* __restrict__ colsum,
        float* __restrict__ colsq) {
    __shared__ float As[16 * AP];

    const int t      = threadIdx.x;
    const int wave   = t >> 5;             // 0..7 -> column tile
    const int lane   = t & 31;
    const int laneLo = lane & 15;
    const int laneHi = lane >> 4;          // 0 or 1
    const int m0 = blockIdx.x * 16;
    const int n0 = wave * 16;
    const int col = n0 + laneLo;

    // ---- preload this wave's B column-slice into registers (32 x v2f) ----
    // B fragment VGPR j holds W[k + j + 2*laneHi][col] == WT[col][k + j + 2*laneHi]
    const float* wcol = WT + (size_t)col * DD + 2 * laneHi;
    v2f breg[32];
    #pragma unroll
    for (int i = 0; i < 32; ++i)
        breg[i] = *(const v2f*)(wcol + i * 4);

    // ---- stage A tile (16 x 128) into LDS with fused construction ----
    {
        const int row = t >> 4;            // 0..15
        const int c0  = (t & 15) << 3;     // 0,8,...,120
        const size_t base = (size_t)(m0 + row) * DD + c0;
        float4 x0 = ((const float4*)(A0 + base))[0];
        float4 x1 = ((const float4*)(A0 + base))[1];
        if (MODE == 0) {
            float4 y0 = ((const float4*)(A1 + base))[0];
            float4 y1 = ((const float4*)(A1 + base))[1];
            x0.x += y0.x; x0.y += y0.y; x0.z += y0.z; x0.w += y0.w;
            x1.x += y1.x; x1.y += y1.y; x1.z += y1.z; x1.w += y1.w;
        } else {
            float4 s0 = *(const float4*)(scale + c0);
            float4 s1 = *(const float4*)(scale + c0 + 4);
            float4 h0 = *(const float4*)(shift + c0);
            float4 h1 = *(const float4*)(shift + c0 + 4);
            x0.x = fmaxf(s0.x * x0.x + h0.x, 0.0f);
            x0.y = fmaxf(s0.y * x0.y + h0.y, 0.0f);
            x0.z = fmaxf(s0.z * x0.z + h0.z, 0.0f);
            x0.w = fmaxf(s0.w * x0.w + h0.w, 0.0f);
            x1.x = fmaxf(s1.x * x1.x + h1.x, 0.0f);
            x1.y = fmaxf(s1.y * x1.y + h1.y, 0.0f);
            x1.z = fmaxf(s1.z * x1.z + h1.z, 0.0f);
            x1.w = fmaxf(s1.w * x1.w + h1.w, 0.0f);
        }
        *(float4*)&As[row * AP + c0]     = x0;
        *(float4*)&As[row * AP + c0 + 4] = x1;
    }
    __syncthreads();

    // ---- WMMA main loop: 128 = 32 steps of K=4 ----
    v8f c = {};
    const float* arow = &As[laneLo * AP + 2 * laneHi];
    #pragma unroll
    for (int i = 0; i < 32; ++i) {
        v2f a = *(const v2f*)(arow + i * 4);
        c = __builtin_amdgcn_wmma_f32_16x16x4_f32(
                /*neg_a=*/false, a, /*neg_b=*/false, breg[i],
                /*c_mod=*/(short)0, c, /*reuse_a=*/false, /*reuse_b=*/false);
    }

    // ---- epilogue: bias, store, column stats ----
    const float bcol = bias[col];
    float s = 0.0f, sq = 0.0f;
    #pragma unroll
    for (int j = 0; j < 8; ++j) {
        float v = c[j] + bcol;
        int row = m0 + j + 8 * laneHi;
        Z[(size_t)row * DD + col] = v;
        s  += v;
        sq += v * v;
    }
    s  += __shfl_down(s, 16, 32);
    sq += __shfl_down(sq, 16, 32);
    if (laneHi == 0) {
        atomicAdd(&colsum[col], s);
        atomicAdd(&colsq[col], sq);
    }
}

// ---------------- BN finalize: mean/var -> affine scale/shift ----------------
__global__ void finalize_stats(const float* __restrict__ colsum,
                               const float* __restrict__ colsq,
                               const float* __restrict__ g,
                               const float* __restrict__ bt,
                               float* __restrict__ scale,
                               float* __restrict__ shift) {
    int d = threadIdx.x;  // 128
    float mean = colsum[d] * (1.0f / (float)NN);
    float var  = colsq[d] * (1.0f / (float)NN) - mean * mean;
    float sc   = g[d] * rsqrtf(var + BN_EPS);
    scale[d] = sc;
    shift[d] = bt[d] - sc * mean;
}

// ---------------- per-node epilogue: BN2+ReLU+residual, then RVQ ----------------
__global__ void post_vq(const float* __restrict__ Z2,
                        const float* __restrict__ scale2,
                        const float* __restrict__ shift2,
                        const float* __restrict__ hin,
                        float* __restrict__ hout,
                        const float* __restrict__ cbn_layer,  // [R, C, D] normalized
                        float* __restrict__ ids_out,          // [N, R*L] as float
                        float* __restrict__ loss,
                        int layer) {
    const int n = blockIdx.x;
    const int d = threadIdx.x;

    __shared__ float sred[DD];
    __shared__ float snorm[DD];
    __shared__ float ssim[CC];
    __shared__ int sidx;

    float z = Z2[(size_t)n * DD + d];
    float v = fmaxf(scale2[d] * z + shift2[d], 0.0f) + hin[(size_t)n * DD + d];
    hout[(size_t)n * DD + d] = v;

    float resid = v;
    for (int r = 0; r < RR; ++r) {
        sred[d] = resid * resid;
        __syncthreads();
        for (int off = 64; off > 0; off >>= 1) {
            if (d < off) sred[d] += sred[d + off];
            __syncthreads();
        }
        float nrm = fmaxf(sqrtf(sred[0]), 1e-12f);
        __syncthreads();
        snorm[d] = resid / nrm;
        __syncthreads();
        if (d < CC) {
            const float* cbrow = cbn_layer + ((size_t)r * CC + d) * DD;
            float dot = 0.0f;
            for (int i = 0; i < DD; ++i) dot += snorm[i] * cbrow[i];
            ssim[d] = dot;
        }
        __syncthreads();
        if (d == 0) {
            int best = 0; float bv = ssim[0];
            for (int cidx = 1; cidx < CC; ++cidx)
                if (ssim[cidx] > bv) { bv = ssim[cidx]; best = cidx; }
            sidx = best;
            ids_out[(size_t)n * (RR * LL) + layer * RR + r] = (float)best;
        }
        __syncthreads();
        float qd = cbn_layer[((size_t)r * CC + sidx) * DD + d];
        if (r == 0) {
            float df = qd - resid;
            sred[d] = df * df;
            __syncthreads();
            for (int off = 64; off > 0; off >>= 1) {
                if (d < off) sred[d] += sred[d + off];
                __syncthreads();
            }
            if (d == 0)
                atomicAdd(loss, COMMIT_W * sred[0] / (float)((long long)NN * DD));
            __syncthreads();
        }
        resid -= qd;
        __syncthreads();
    }
}

// ---------------- host-side launch ----------------

extern "C" void kernel_launch(void* const* d_in, const int* in_sizes, int n_in,
                              void* d_out, int out_size, void* d_ws, size_t ws_size,
                              hipStream_t stream) {
    (void)in_sizes; (void)n_in; (void)out_size; (void)ws_size;

    const float* x   = (const float*)d_in[0];
    const int*   src = (const int*)d_in[1];
    const int*   dst = (const int*)d_in[2];
    const float* W1  = (const float*)d_in[3];
    const float* b1  = (const float*)d_in[4];
    const float* g1  = (const float*)d_in[5];
    const float* bt1 = (const float*)d_in[6];
    const float* W2  = (const float*)d_in[7];
    const float* b2  = (const float*)d_in[8];
    const float* g2  = (const float*)d_in[9];
    const float* bt2 = (const float*)d_in[10];
    const float* cb  = (const float*)d_in[11];

    const long long ND = (long long)NN * DD;

    float* out   = (float*)d_out;
    float* hout  = out;              // [N, D]
    float* loss  = out + ND;         // [1]
    float* ids   = out + ND + 1;     // [N, R*L]

    float* ws     = (float*)d_ws;
    float* neigh  = ws;                    // [N, D]  (reused as z2)
    float* z1     = ws + ND;               // [N, D]
    float* colsum = ws + 2 * ND;           // [128]
    float* colsq  = colsum + DD;           // [128]
    float* scale  = colsum + 2 * DD;       // [128]
    float* shift  = colsum + 3 * DD;       // [128]
    float* cbn    = colsum + 4 * DD;       // [L*R*C, D]
    float* WT1    = cbn + (size_t)LL * RR * CC * DD;   // [L, D, D] transposed
    float* WT2    = WT1 + (size_t)LL * DD * DD;        // [L, D, D] transposed

    // init: loss = 0, normalize codebooks, transpose weights
    zero_f32<<<1, 32, 0, stream>>>(loss, 1);
    norm_codebooks<<<LL * RR * CC, DD, 0, stream>>>(cb, cbn);
    {
        dim3 tg((DD * DD) / 256, LL);
        transpose_w<<<tg, 256, 0, stream>>>(W1, WT1);
        transpose_w<<<tg, 256, 0, stream>>>(W2, WT2);
    }

    const int gemm_grid    = NN / 16;            // 3125
    const int scatter_grid = (EE * DD) / 256;    // 400000
    const int zero_grid    = (int)((ND + 255) / 256);

    for (int l = 0; l < LL; ++l) {
        const float* h = (l == 0) ? x : hout;

        // GIN aggregation
        zero_f32<<<zero_grid, 256, 0, stream>>>(neigh, ND);
        scatter_add_kernel<<<scatter_grid, 256, 0, stream>>>(h, src, dst, neigh);

        // Linear1 (A = h + neigh), gather BN1 stats
        zero_f32<<<1, 256, 0, stream>>>(colsum, 2 * DD);
        gemm_bn_stats<0><<<gemm_grid, 256, 0, stream>>>(
            h, neigh, nullptr, nullptr,
            WT1 + (size_t)l * DD * DD, b1 + (size_t)l * DD,
            z1, colsum, colsq);
        finalize_stats<<<1, DD, 0, stream>>>(colsum, colsq,
                                             g1 + (size_t)l * DD, bt1 + (size_t)l * DD,
                                             scale, shift);

        // Linear2 (A = relu(BN1(z1))), gather BN2 stats; z2 overwrites neigh
        zero_f32<<<1, 256, 0, stream>>>(colsum, 2 * DD);
        gemm_bn_stats<1><<<gemm_grid, 256, 0, stream>>>(
            z1, nullptr, scale, shift,
            WT2 + (size_t)l * DD * DD, b2 + (size_t)l * DD,
            neigh, colsum, colsq);
        finalize_stats<<<1, DD, 0, stream>>>(colsum, colsq,
                                             g2 + (size_t)l * DD, bt2 + (size_t)l * DD,
                                             scale, shift);

        // BN2 + ReLU + residual + 3-stage RVQ, write new h into d_out
        post_vq<<<NN, DD, 0, stream>>>(neigh, scale, shift, h, hout,
                                       cbn + (size_t)l * RR * CC * DD,
                                       ids, loss, l);
    }
}